// ImplicitFunction_73031623901887
// MI455X (gfx1250) — compile-verified
//
#include <hip/hip_runtime.h>

typedef __attribute__((ext_vector_type(2))) float v2f;
typedef __attribute__((ext_vector_type(8))) float v8f;

#define L_HIDDEN     6
#define NEG_SLOPE    0.2f
#define XSTRIDE      68                 // padded activation row stride (floats)
#define WFRAG_LAYER  (64 * 64)          // 4096 floats/layer, B-fragment-swizzled (no padding needed)

// Async global->LDS staging if the toolchain declares the gfx1250 builtins.
#if defined(__AMDGCN__) && __has_builtin(__builtin_amdgcn_global_load_async_to_lds_b32) && \
    __has_builtin(__builtin_amdgcn_s_wait_asynccnt)
#define USE_ASYNC_LDS 1
#else
#define USE_ASYNC_LDS 0
#endif

typedef __attribute__((address_space(1))) int glb_i32;
typedef __attribute__((address_space(3))) int lds_i32;

// B-fragment swizzle: raw wh element (i, r, c)  [r = input dim (K), c = output dim (N)]
// -> lane (hi=(r>>1)&1, lo=c&15), pair elem j=r&1, chunk kc=r>>2, tile tt=c>>4.
__device__ __forceinline__ int wfrag_index(int i, int r, int c) {
    int kc = r >> 2, j = r & 1, hi = (r >> 1) & 1;
    int tt = c >> 4, lo = c & 15;
    return ((i * 64 + tt * 16 + kc) * 32 + hi * 16 + lo) * 2 + j;
}

__global__ __launch_bounds__(256)
void implicit_mlp_kernel(const float* __restrict__ pts,
                         const float* __restrict__ w0,
                         const float* __restrict__ s0,
                         const float* __restrict__ b0,
                         const float* __restrict__ wh,
                         const float* __restrict__ sh,
                         const float* __restrict__ bh,
                         const float* __restrict__ wo,
                         const float* __restrict__ so,
                         const float* __restrict__ bo,
                         float* __restrict__ out,
                         int npts)
{
    __shared__ float s_wb[L_HIDDEN * WFRAG_LAYER];   // 98,304 B  B-fragments
    __shared__ float s_x[8 * 16 * XSTRIDE];          // 34,816 B  8 waves x 16x64 tile
    __shared__ float s_w0[192];
    __shared__ float s_s0[64], s_b0[64], s_wo[64];
    __shared__ float s_sh[L_HIDDEN * 64], s_bh[L_HIDDEN * 64];

    const int t = threadIdx.x;

    // ---- stage hidden-layer weights into LDS in B-fragment order ----
#if USE_ASYNC_LDS
    for (int e = t; e < L_HIDDEN * 64 * 64; e += 256) {
        int i = e >> 12, r = (e >> 6) & 63, c = e & 63;
        int d = wfrag_index(i, r, c);
        __builtin_amdgcn_global_load_async_to_lds_b32(
            (glb_i32*)(wh + e),
            (lds_i32*)&s_wb[d],
            /*offset=*/0, /*cpol=*/0);
    }
#else
    for (int e = t; e < L_HIDDEN * 64 * 64; e += 256) {
        int i = e >> 12, r = (e >> 6) & 63, c = e & 63;
        s_wb[wfrag_index(i, r, c)] = wh[e];
    }
#endif
    // ---- small params: plain staging ----
    for (int e = t; e < 192; e += 256) s_w0[e] = w0[e];
    for (int e = t; e < 64; e += 256) { s_s0[e] = s0[e]; s_b0[e] = b0[e]; s_wo[e] = wo[e]; }
    for (int e = t; e < L_HIDDEN * 64; e += 256) { s_sh[e] = sh[e]; s_bh[e] = bh[e]; }
#if USE_ASYNC_LDS
    __builtin_amdgcn_s_wait_asynccnt(0);
#endif
    __syncthreads();

    const int wv   = t >> 5;
    const int lane = t & 31;
    const int lo   = lane & 15;
    const int hi   = lane >> 4;
    const int g    = blockIdx.x * 128 + wv * 16 + lo;
    const int gc   = (g < npts) ? g : (npts - 1);
    float* xw = &s_x[wv * 16 * XSTRIDE];

    // ---- layer 0: [3 -> 64] ----
    {
        float p0 = pts[3 * gc + 0], p1 = pts[3 * gc + 1], p2 = pts[3 * gc + 2];
        #pragma unroll
        for (int kk = 0; kk < 32; ++kk) {
            int k = hi * 32 + kk;
            float v = p0 * s_w0[k] + p1 * s_w0[64 + k] + p2 * s_w0[128 + k];
            v = v * s_s0[k] + s_b0[k];
            xw[lo * XSTRIDE + k] = (v >= 0.f) ? v : NEG_SLOPE * v;
        }
    }

    // ---- 6 hidden layers: Y[16x64] = X[16x64] * W[64x64], fp32 WMMA 16x16x4 ----
    for (int i = 0; i < L_HIDDEN; ++i) {
        __syncthreads();
        const float* WB = &s_wb[i * WFRAG_LAYER];

        v2f A[16];
        #pragma unroll
        for (int kc = 0; kc < 16; ++kc)
            A[kc] = *(const v2f*)&xw[lo * XSTRIDE + kc * 4 + 2 * hi];

        #pragma unroll
        for (int tt = 0; tt < 4; ++tt) {
            v8f C = {};
            #pragma unroll
            for (int kc = 0; kc < 16; ++kc) {
                v2f B = *(const v2f*)&WB[((tt * 16 + kc) * 32 + lane) * 2];  // single ds_load_b64
                C = __builtin_amdgcn_wmma_f32_16x16x4_f32(
                        false, A[kc], false, B, (short)0, C, false, false);
            }
            float s = s_sh[i * 64 + tt * 16 + lo];
            float b = s_bh[i * 64 + tt * 16 + lo];
            #pragma unroll
            for (int j = 0; j < 8; ++j) {
                float y = C[j] * s + b;
                y = (y >= 0.f) ? y : NEG_SLOPE * y;
                xw[(j + 8 * hi) * XSTRIDE + tt * 16 + lo] = y;   // A regs already loaded; safe
            }
        }
    }
    __syncthreads();

    // ---- output layer: [64 -> 1] ----
    if (hi == 0 && g < npts) {
        float acc = 0.f;
        #pragma unroll
        for (int k = 0; k < 64; k += 4) {
            float4 xv  = *(const float4*)&xw[lo * XSTRIDE + k];
            float4 wv4 = *(const float4*)&s_wo[k];
            acc += xv.x * wv4.x + xv.y * wv4.y + xv.z * wv4.z + xv.w * wv4.w;
        }
        out[g] = acc * so[0] + bo[0];
    }
}

extern "C" void kernel_launch(void* const* d_in, const int* in_sizes, int n_in,
                              void* d_out, int out_size, void* d_ws, size_t ws_size,
                              hipStream_t stream) {
    (void)n_in; (void)d_ws; (void)ws_size; (void)out_size;
    const float* pts = (const float*)d_in[0];
    const float* w0  = (const float*)d_in[1];
    const float* s0  = (const float*)d_in[2];
    const float* b0  = (const float*)d_in[3];
    const float* wh  = (const float*)d_in[4];
    const float* sh  = (const float*)d_in[5];
    const float* bh  = (const float*)d_in[6];
    const float* wo  = (const float*)d_in[7];
    const float* so  = (const float*)d_in[8];
    const float* bo  = (const float*)d_in[9];
    float* out = (float*)d_out;

    int npts = in_sizes[0] / 3;
    int nblocks = (npts + 127) / 128;
    implicit_mlp_kernel<<<dim3(nblocks), dim3(256), 0, stream>>>(
        pts, w0, s0, b0, wh, sh, bh, wo, so, bo, out, npts);
}